// L1OrderLoss_64733747085527
// MI455X (gfx1250) — compile-verified
//
#include <hip/hip_runtime.h>

// Chamfer-style L1 order loss for [8192,128] f32 pred/target, PTS_DIM=2.
// One wave32 per row: stage pred/target rows (512 B each) into LDS via the
// CDNA5 async global->LDS DMA path, then each lane owns 2 pred points and
// 2 target points (4 coords each) and brute-forces the 64-long min scans
// with broadcast LDS reads. Deterministic 2-stage reduction to a scalar.

typedef int v4i __attribute__((vector_size(16)));
typedef __attribute__((address_space(1))) v4i* as1_v4i_ptr;
typedef __attribute__((address_space(3))) v4i* as3_v4i_ptr;

__device__ __forceinline__ float wave_reduce_sum(float v) {
  v += __shfl_xor(v, 16, 32);
  v += __shfl_xor(v,  8, 32);
  v += __shfl_xor(v,  4, 32);
  v += __shfl_xor(v,  2, 32);
  v += __shfl_xor(v,  1, 32);
  return v;
}

__global__ __launch_bounds__(256) void chamfer_rows_kernel(
    const float* __restrict__ pred, const float* __restrict__ target,
    float* __restrict__ partials, int nrows) {
  __shared__ float rowbuf[8][256];  // per wave: [0..127]=pred row, [128..255]=target row
  __shared__ float wsum[8];

  const int wave = threadIdx.x >> 5;
  const int lane = threadIdx.x & 31;
  const int row  = blockIdx.x * 8 + wave;  // 8 waves per block, 1 row per wave

  float* lp = &rowbuf[wave][0];
  float* lt = &rowbuf[wave][128];
  float partial = 0.0f;

  if (row < nrows) {
    const float* prow = pred   + (size_t)row * 128;
    const float* trow = target + (size_t)row * 128;

#if __has_builtin(__builtin_amdgcn_global_load_async_to_lds_b128)
    // CDNA5 async DMA: 32 lanes x 16B = one full 512B row per instruction.
    __builtin_amdgcn_global_load_async_to_lds_b128(
        (as1_v4i_ptr)(unsigned long long)(prow + lane * 4),
        (as3_v4i_ptr)(lp + lane * 4), 0, 0);
    __builtin_amdgcn_global_load_async_to_lds_b128(
        (as1_v4i_ptr)(unsigned long long)(trow + lane * 4),
        (as3_v4i_ptr)(lt + lane * 4), 0, 0);
#if __has_builtin(__builtin_amdgcn_s_wait_asynccnt)
    __builtin_amdgcn_s_wait_asynccnt(0);
#else
    asm volatile("s_wait_asynccnt 0" ::: "memory");
#endif
#else
    // Fallback: VGPR-staged copy into LDS (same-wave LDS ops are in-order).
    const float4 pv = ((const float4*)prow)[lane];
    const float4 tv = ((const float4*)trow)[lane];
    ((float4*)lp)[lane] = pv;
    ((float4*)lt)[lane] = tv;
#endif

    // Lane owns pred points {lane, lane+32} and target points {lane, lane+32}.
    const float p0x = lp[2 * lane + 0],  p0y = lp[2 * lane + 1];
    const float p1x = lp[2 * lane + 64], p1y = lp[2 * lane + 65];
    const float t0x = lt[2 * lane + 0],  t0y = lt[2 * lane + 1];
    const float t1x = lt[2 * lane + 64], t1y = lt[2 * lane + 65];

    float a0 = 1e30f, a1 = 1e30f, a2 = 1e30f, a3 = 1e30f;  // dist_1 mins
    float b0 = 1e30f, b1 = 1e30f, b2 = 1e30f, b3 = 1e30f;  // dist_2 mins
#pragma unroll 8
    for (int j = 0; j < 64; ++j) {
      // Uniform-address LDS reads -> hardware broadcast.
      const float tx = lt[2 * j], ty = lt[2 * j + 1];
      a0 = fminf(a0, fabsf(p0x - tx));
      a1 = fminf(a1, fabsf(p0y - ty));
      a2 = fminf(a2, fabsf(p1x - tx));
      a3 = fminf(a3, fabsf(p1y - ty));
      const float px = lp[2 * j], py = lp[2 * j + 1];
      b0 = fminf(b0, fabsf(t0x - px));
      b1 = fminf(b1, fabsf(t0y - py));
      b2 = fminf(b2, fabsf(t1x - px));
      b3 = fminf(b3, fabsf(t1y - py));
    }
    partial = ((a0 + a1) + (a2 + a3)) + ((b0 + b1) + (b2 + b3));
  }

  const float ws = wave_reduce_sum(partial);
  if (lane == 0) wsum[wave] = ws;
  __syncthreads();
  if (threadIdx.x == 0) {
    float s = 0.0f;
#pragma unroll
    for (int w = 0; w < 8; ++w) s += wsum[w];
    partials[blockIdx.x] = s;
  }
}

__global__ __launch_bounds__(256) void reduce_partials_kernel(
    const float* __restrict__ partials, float* __restrict__ out,
    int n, float scale) {
  __shared__ float wsum[8];
  float v = 0.0f;
  for (int i = threadIdx.x; i < n; i += 256) v += partials[i];
  v = wave_reduce_sum(v);
  if ((threadIdx.x & 31) == 0) wsum[threadIdx.x >> 5] = v;
  __syncthreads();
  if (threadIdx.x == 0) {
    float s = 0.0f;
#pragma unroll
    for (int w = 0; w < 8; ++w) s += wsum[w];
    out[0] = s * scale;  // 0.5 * mean / num_pts
  }
}

extern "C" void kernel_launch(void* const* d_in, const int* in_sizes, int n_in,
                              void* d_out, int out_size, void* d_ws, size_t ws_size,
                              hipStream_t stream) {
  const float* pred   = (const float*)d_in[0];
  const float* target = (const float*)d_in[1];
  const int nrows   = in_sizes[0] / 128;   // 8192
  const int nblocks = (nrows + 7) / 8;     // 1024 (8 waves/block, 1 row/wave)
  float* partials = (float*)d_ws;          // nblocks floats of scratch

  chamfer_rows_kernel<<<nblocks, 256, 0, stream>>>(pred, target, partials, nrows);

  // out = 0.5 * S / (nrows * 128 * 64); exact power-of-two scale for 8192 rows.
  const float scale = 0.5f / ((float)nrows * 128.0f * 64.0f);
  reduce_partials_kernel<<<1, 256, 0, stream>>>(partials, (float*)d_out, nblocks, scale);
}